// KANLayer_41214506172896
// MI455X (gfx1250) — compile-verified
//
#include <hip/hip_runtime.h>

typedef __attribute__((ext_vector_type(2))) float v2f;
typedef __attribute__((ext_vector_type(4))) float v4f;
typedef __attribute__((ext_vector_type(8))) float v8f;

#define IN_F   512
#define OUT_F  512
#define NUM_CPT 16
#define DEG    3
#define NB     13               // NUM_CP - DEGREE
#define BATCH  4096
#define KDIM   (IN_F * NB)      // 6656
#define RMIN  (-3.0f)
#define RMAX   (3.0f)
#define BEPS   1.1920929e-5f    // FLT_EPSILON * 100

// Async global->LDS path only exists / assembles in the gfx1250 device pass.
#if defined(__gfx1250__)
#define HAVE_ASYNC 1
#else
#define HAVE_ASYNC 0
#endif

#if HAVE_ASYNC
// GLOBAL_LOAD_ASYNC_TO_LDS_B128: vdst = LDS byte-address VGPR, vaddr = 64-bit
// global address (VGPR pair), saddr = off. Tracked by ASYNCcnt.
__device__ __forceinline__ void async_cp16(const void* g, const void* l) {
  unsigned lds = (unsigned)(uintptr_t)l;   // flat LDS aperture: offset in addr[31:0]
  asm volatile("global_load_async_to_lds_b128 %0, %1, off"
               :: "v"(lds), "v"(g) : "memory");
}
__device__ __forceinline__ void wait_async0() {
  asm volatile("s_wait_asynccnt 0" ::: "memory");
}
#endif

// knots: [0,0,0, linspace(0,1,14), 1,1,1] -> knot(j) = clamp((j-3)/13, 0, 1)
__device__ __forceinline__ float knotf(int j) {
  float v = (float)(j - 3) * (1.0f / 13.0f);
  return v < 0.0f ? 0.0f : (v > 1.0f ? 1.0f : v);
}

// Cox–de Boor, degree 3, 16 control points -> 13 basis values (constants fold).
__device__ __forceinline__ void bspline13(float xs, float* Nv) {
  float N[NUM_CPT];
#pragma unroll
  for (int j = 0; j < NUM_CPT; ++j) {
    float lo = knotf(j), hi = knotf(j + 1);
    N[j] = (xs >= lo && xs < hi) ? 1.0f : 0.0f;
  }
  if (xs == 1.0f) N[NUM_CPT - 1] += 1.0f;
#pragma unroll
  for (int d = 1; d <= DEG; ++d) {
#pragma unroll
    for (int j = 0; j < NUM_CPT - 1; ++j) {
      if (j < NUM_CPT - d) {
        float ki   = knotf(j);
        float kid  = knotf(j + d);
        float d1   = kid - ki;   d1 = d1 > BEPS ? d1 : BEPS;
        float ki1  = knotf(j + 1);
        float kid1 = knotf(j + d + 1);
        float d2   = kid1 - ki1; d2 = d2 > BEPS ? d2 : BEPS;
        N[j] = (xs - ki) * (1.0f / d1) * N[j] + (kid1 - xs) * (1.0f / d2) * N[j + 1];
      }
    }
  }
#pragma unroll
  for (int j = 0; j < NB; ++j) Nv[j] = N[j];
}

// ---------------- stage 1: A[b][i*13+k] = basis(x[b][i]) ----------------
__global__ void __launch_bounds__(256)
basis_stage_kernel(const float* __restrict__ x, float* __restrict__ A) {
  int idx = blockIdx.x * 256 + threadIdx.x;      // b*512 + i
  if (idx >= BATCH * IN_F) return;
  float xs = (x[idx] - RMIN) * (1.0f / (RMAX - RMIN));
  xs = xs < 0.0f ? 0.0f : (xs > 1.0f ? 1.0f : xs);
  float Nv[NB];
  bspline13(xs, Nv);
  float* dst = A + (size_t)idx * NB;             // == b*6656 + i*13
#pragma unroll
  for (int j = 0; j < NB; ++j) dst[j] = Nv[j];
}

// -------- stage 2: Wt pair-interleaved: Wt[(k>>1)*1024 + 2*o + (k&1)] = cp[i][o][kb]
// (k = i*13+kb).  A 16B packet of Wt = {(k,n),(k+1,n),(k,n+1),(k+1,n+1)} so the
// async B128 copy lands LDS in fragment order: (b.x,b.y) contiguous -> 1x ds_load_b64.
__global__ void __launch_bounds__(256)
cp_transpose_kernel(const float* __restrict__ cp, float* __restrict__ Wt) {
  int idx = blockIdx.x * 256 + threadIdx.x;      // i*512 + o
  if (idx >= IN_F * OUT_F) return;
  int i = idx >> 9, o = idx & 511;
  const float* src = cp + (size_t)idx * NB;
#pragma unroll
  for (int kb = 0; kb < NB; ++kb) {
    int k = i * NB + kb;
    Wt[(size_t)(k >> 1) * (2 * OUT_F) + 2 * o + (k & 1)] = src[kb];
  }
}

// ---------------- stage 3: FP32 WMMA GEMM  out[4096,512] = A @ W ----------------
#define BM 128
#define BN 128
#define KC 32
#define LDA 36    // sA m-major row stride (floats): conflict-free b64 frags, 16B rows
#define LDP 288   // sB pair-row stride (floats): 288 % 64 == 32 -> halves use disjoint banks
#define NCHUNK (KDIM / KC)   // 208

__global__ void __launch_bounds__(256)
kan_wmma_gemm(const float* __restrict__ A, const float* __restrict__ Wt,
              float* __restrict__ out) {
  __shared__ float sA[2][BM * LDA];           // 18,432 B each
  __shared__ float sB[2][(KC / 2) * LDP];     // 18,432 B each (16 pair-rows)

  const int tid  = threadIdx.x;
  const int wave = tid >> 5;
  const int lane = tid & 31;
  const int half = lane >> 4;
  const int l16  = lane & 15;
  const int wm   = (wave & 3) * 32;    // 4 waves tile M
  const int wn   = (wave >> 2) * 64;   // 2 waves tile N, 64 wide each
  const int b0   = blockIdx.y * BM;
  const int n0   = blockIdx.x * BN;

  // copy thread mapping (per 16B packet)
  const int ar = tid >> 3, ac = (tid & 7) * 4;     // A: 32 rows x 32k per iter
  const int bp = tid >> 6, bc = (tid & 63) * 4;    // B: 4 pair-rows x 256 floats per iter

  v8f acc[2][4] = {};

#if HAVE_ASYNC
  auto gcopy = [&](int c, int buf) {
    const int k0 = c * KC;
    const int p0 = k0 >> 1;
#pragma unroll
    for (int it = 0; it < 4; ++it)
      async_cp16(A + (size_t)(b0 + it * 32 + ar) * KDIM + k0 + ac,
                 &sA[buf][(it * 32 + ar) * LDA + ac]);
#pragma unroll
    for (int it = 0; it < 4; ++it)
      async_cp16(Wt + (size_t)(p0 + it * 4 + bp) * (2 * OUT_F) + 2 * n0 + bc,
                 &sB[buf][(it * 4 + bp) * LDP + bc]);
  };
#else
  v4f ra[4], rb[4];
  auto gload = [&](int c) {
    const int k0 = c * KC;
    const int p0 = k0 >> 1;
#pragma unroll
    for (int it = 0; it < 4; ++it)
      ra[it] = *(const v4f*)(A + (size_t)(b0 + it * 32 + ar) * KDIM + k0 + ac);
#pragma unroll
    for (int it = 0; it < 4; ++it)
      rb[it] = *(const v4f*)(Wt + (size_t)(p0 + it * 4 + bp) * (2 * OUT_F) + 2 * n0 + bc);
  };
  auto sstore = [&](int buf) {
#pragma unroll
    for (int it = 0; it < 4; ++it)
      *(v4f*)(&sA[buf][(it * 32 + ar) * LDA + ac]) = ra[it];
#pragma unroll
    for (int it = 0; it < 4; ++it)
      *(v4f*)(&sB[buf][(it * 4 + bp) * LDP + bc]) = rb[it];
  };
#endif

  auto compute = [&](int buf) {
    const float* pa = &sA[buf][0];
    const float* pb = &sB[buf][0];
#pragma unroll
    for (int kk = 0; kk < KC; kk += 4) {
      v2f af[2], bf[4];
#pragma unroll
      for (int s = 0; s < 2; ++s)   // A 16x4 frag: lanes0-15 K0/K1, lanes16-31 K2/K3
        af[s] = *(const v2f*)(pa + (wm + s * 16 + l16) * LDA + kk + half * 2);
#pragma unroll
      for (int t = 0; t < 4; ++t)   // B frag: pair-interleaved -> one aligned b64
        bf[t] = *(const v2f*)(pb + ((kk >> 1) + half) * LDP + 2 * (wn + t * 16 + l16));
#pragma unroll
      for (int s = 0; s < 2; ++s)
#pragma unroll
        for (int t = 0; t < 4; ++t)
          acc[s][t] = __builtin_amdgcn_wmma_f32_16x16x4_f32(
              false, af[s], false, bf[t], (short)0, acc[s][t], false, false);
    }
  };

#if HAVE_ASYNC
  gcopy(0, 0);
  wait_async0();
  __syncthreads();
  int buf = 0;
  for (int c = 0; c < NCHUNK; ++c) {
    if (c + 1 < NCHUNK) gcopy(c + 1, buf ^ 1);  // DMA into other buffer
    compute(buf);
    if (c + 1 < NCHUNK) wait_async0();          // own async writes done
    __syncthreads();                            // single barrier: reads + DMA visible
    buf ^= 1;
  }
#else
  gload(0);
  sstore(0);
  __syncthreads();
  int buf = 0;
  for (int c = 0; c < NCHUNK; ++c) {
    if (c + 1 < NCHUNK) gload(c + 1);
    compute(buf);
    __syncthreads();
    if (c + 1 < NCHUNK) sstore(buf ^ 1);
    __syncthreads();
    buf ^= 1;
  }
#endif

  // C/D layout: VGPR v -> row v (lanes 0-15) / row v+8 (lanes 16-31), col = lane%16
#pragma unroll
  for (int s = 0; s < 2; ++s) {
    const int mrow = b0 + wm + s * 16 + half * 8;
#pragma unroll
    for (int t = 0; t < 4; ++t) {
      const int ncol = n0 + wn + t * 16 + l16;
#pragma unroll
      for (int v = 0; v < 8; ++v)
        out[(size_t)(mrow + v) * OUT_F + ncol] = acc[s][t][v];
    }
  }
}

// ---------------- fallback (workspace too small): LDS-shared basis ----------------
__global__ void __launch_bounds__(512)
kan_fallback_kernel(const float* __restrict__ x, const float* __restrict__ cp,
                    float* __restrict__ out) {
  __shared__ float sN[IN_F * NB];
  const int b = blockIdx.x;
  const int tid = threadIdx.x;
  {
    float xs = (x[(size_t)b * IN_F + tid] - RMIN) * (1.0f / (RMAX - RMIN));
    xs = xs < 0.0f ? 0.0f : (xs > 1.0f ? 1.0f : xs);
    float Nv[NB];
    bspline13(xs, Nv);
#pragma unroll
    for (int j = 0; j < NB; ++j) sN[tid * NB + j] = Nv[j];
  }
  __syncthreads();
  float acc = 0.0f;
  for (int i = 0; i < IN_F; ++i) {
    const float* c = cp + ((size_t)i * OUT_F + tid) * NB;
#pragma unroll
    for (int j = 0; j < NB; ++j) acc += sN[i * NB + j] * c[j];
  }
  out[(size_t)b * OUT_F + tid] = acc;
}

extern "C" void kernel_launch(void* const* d_in, const int* in_sizes, int n_in,
                              void* d_out, int out_size, void* d_ws, size_t ws_size,
                              hipStream_t stream) {
  const float* x  = (const float*)d_in[0];
  const float* cp = (const float*)d_in[1];
  float* out = (float*)d_out;

  const size_t A_BYTES = (size_t)BATCH * KDIM * sizeof(float);   // 109.1 MB
  const size_t W_BYTES = (size_t)KDIM * OUT_F * sizeof(float);   //  13.6 MB

  if (ws_size >= A_BYTES + W_BYTES) {
    float* Aws = (float*)d_ws;
    float* Wt  = (float*)((char*)d_ws + A_BYTES);
    basis_stage_kernel<<<(BATCH * IN_F + 255) / 256, 256, 0, stream>>>(x, Aws);
    cp_transpose_kernel<<<(IN_F * OUT_F + 255) / 256, 256, 0, stream>>>(cp, Wt);
    kan_wmma_gemm<<<dim3(OUT_F / BN, BATCH / BM), 256, 0, stream>>>(Aws, Wt, out);
  } else {
    kan_fallback_kernel<<<BATCH, 512, 0, stream>>>(x, cp, out);
  }
}